// CapsNet_6957847020105
// MI455X (gfx1250) — compile-verified
//
#include <hip/hip_runtime.h>
#include <math.h>

// ---------------- model constants ----------------
#define BATCH 64
#define A_CH 64
#define B_CAPS 8
#define C_CAPS 16
#define D_CAPS 16
#define E_CLS 10
#define PP 16            // PSIZE = P*P
#define KK 3
#define NI 72            // K*K*B = 3*3*8
#define OH 6             // conv_caps output spatial
#define NCAP 576         // 6*6*16 class-caps input capsules
#define EPSF 1e-8f
#define LAMF 1e-3f
#define LN2PI 1.8378770664093453f

typedef float v2f __attribute__((ext_vector_type(2)));
typedef float v8f __attribute__((ext_vector_type(8)));

// ---------------------------------------------------------------
// 1) conv1 5x5 stride2 pad2 + ReLU : x(64,1,28,28) -> h(64,64,14,14) NCHW
// ---------------------------------------------------------------
__global__ void conv1_relu(const float* __restrict__ x, const float* __restrict__ w,
                           const float* __restrict__ bias, float* __restrict__ h) {
  int t = blockIdx.x * blockDim.x + threadIdx.x;
  if (t >= BATCH * A_CH * 14 * 14) return;
  int xo = t % 14, yo = (t / 14) % 14, c = (t / 196) % A_CH, b = t / (A_CH * 196);
  float acc = bias[c];
  for (int ky = 0; ky < 5; ++ky) {
    int iy = 2 * yo - 2 + ky;
    if (iy < 0 || iy >= 28) continue;
    for (int kx = 0; kx < 5; ++kx) {
      int ix = 2 * xo - 2 + kx;
      if (ix < 0 || ix >= 28) continue;
      acc += x[(b * 28 + iy) * 28 + ix] * w[(c * 5 + ky) * 5 + kx];
    }
  }
  h[t] = fmaxf(acc, 0.0f);
}

// ---------------------------------------------------------------
// 2) pc_pose 1x1 conv as WMMA f32 GEMM:
//    pose[m,o] = sum_k h[m,k]*W[o,k] + b[o]; m=12544 (b*196+s), K=64, N=128
//    A tile 16x4 f32 layout: lanes 0-15 K0/K1, lanes 16-31 K2/K3 (ISA 7.12.2)
// ---------------------------------------------------------------
__global__ void pc_pose_wmma(const float* __restrict__ h, const float* __restrict__ w,
                             const float* __restrict__ bias, float* __restrict__ pose) {
  int nt = blockIdx.x;            // 0..7   (N=128)
  int mt = blockIdx.y;            // 0..783 (M=12544)
  int lane = threadIdx.x;         // wave32
  int hf = lane >> 4;             // half: 0 => K0/K1, 1 => K2/K3
  int l  = lane & 15;
  int m = mt * 16 + l;
  int bidx = m / 196, s = m % 196;
  int o = nt * 16 + l;
  v8f acc = {};
  for (int k0 = 0; k0 < 64; k0 += 4) {
    int ka = k0 + hf * 2;
    v2f av, bv;
    const float* hp = &h[bidx * 12544 + ka * 196 + s];   // NCHW gather
    av.x = hp[0];  av.y = hp[196];
    bv.x = w[o * 64 + ka]; bv.y = w[o * 64 + ka + 1];
    acc = __builtin_amdgcn_wmma_f32_16x16x4_f32(false, av, false, bv,
                                                (short)0, acc, false, false);
  }
  int col = nt * 16 + l;
  float bb = bias[col];
  for (int v = 0; v < 8; ++v) {
    int row = mt * 16 + v + hf * 8;                      // D layout: M=v (+8 hi half)
    pose[row * 128 + col] = acc[v] + bb;                 // NHWC output
  }
}

// ---------------------------------------------------------------
// 3) pc_a 1x1 conv + sigmoid : act NHWC (b,14,14,8)
// ---------------------------------------------------------------
__global__ void pc_act(const float* __restrict__ h, const float* __restrict__ w,
                       const float* __restrict__ bias, float* __restrict__ act) {
  int t = blockIdx.x * blockDim.x + threadIdx.x;
  if (t >= 12544 * 8) return;
  int j = t & 7, m = t >> 3;
  int b = m / 196, s = m % 196;
  float acc = bias[j];
  for (int k = 0; k < 64; ++k) acc += h[b * 12544 + k * 196 + s] * w[j * 64 + k];
  act[m * 8 + j] = 1.0f / (1.0f + __expf(-acc));
}

// ---------------------------------------------------------------
// 4) conv_caps: votes + 3-iter EM routing fully fused, one block per
//    output capsule position n (2304 blocks). Votes live in LDS (72KB).
// ---------------------------------------------------------------
__global__ void conv_caps_em(const float* __restrict__ pose, const float* __restrict__ act,
                             const float* __restrict__ ccw,
                             const float* __restrict__ beta_u, const float* __restrict__ beta_a,
                             float* __restrict__ pose1, float* __restrict__ a1) {
  extern __shared__ float sm[];
  float* V    = sm;                 // 72*16*16
  float* R    = V + NI * 256;       // 72*16
  float* sA   = R + NI * 16;        // 72
  float* rsum = sA + NI;            // 16
  float* MU   = rsum + 16;          // 256
  float* SG   = MU + 256;           // 256
  float* AO   = SG + 256;           // 16

  int n = blockIdx.x;
  int b = n / 36, s = n % 36, oy = s / 6, ox = s % 6;
  int tid = threadIdx.x;

  if (tid < NI) {
    int ky = tid / 24, kx = (tid / 8) % 3, bb = tid & 7;
    sA[tid] = act[((b * 14 + (oy * 2 + ky)) * 14 + (ox * 2 + kx)) * 8 + bb];
  }
  for (int e = tid; e < NI * 256; e += 256) {
    int i = e >> 8, rest = e & 255, c = rest >> 4, hh = rest & 15;
    int p = hh >> 2, rr = hh & 3;
    int ky = i / 24, kx = (i / 8) % 3, bb = i & 7;
    const float* ps = &pose[((b * 14 + (oy * 2 + ky)) * 14 + (ox * 2 + kx)) * 128 + bb * 16 + p * 4];
    const float* wv = &ccw[(i * 16 + c) * 16 + rr];
    float acc = ps[0] * wv[0] + ps[1] * wv[4] + ps[2] * wv[8] + ps[3] * wv[12];
    V[e] = acc;
    if (e < NI * 16) R[e] = 1.0f / 16.0f;
  }
  __syncthreads();

  for (int it = 0; it < 3; ++it) {
    if (tid < NI) {                          // rr = r*a_in, normalize over c
      float sum = 0.f;
      for (int c = 0; c < 16; ++c) { float t = R[tid * 16 + c] * sA[tid]; R[tid * 16 + c] = t; sum += t; }
      float inv = 1.0f / (sum + EPSF);
      for (int c = 0; c < 16; ++c) R[tid * 16 + c] *= inv;
    }
    __syncthreads();
    if (tid < 16) {                          // r_sum per c
      float sum = 0.f;
      for (int i = 0; i < NI; ++i) sum += R[i * 16 + tid];
      rsum[tid] = sum;
    }
    __syncthreads();
    {                                        // mu: 256 outputs, one per thread
      int c = tid >> 4, hh = tid & 15;
      float num = 0.f;
      for (int i = 0; i < NI; ++i) num += R[i * 16 + c] * V[(i * 16 + c) * 16 + hh];
      MU[tid] = num / (rsum[c] + EPSF);
    }
    __syncthreads();
    {                                        // sigma_sq
      int c = tid >> 4, hh = tid & 15;
      float m = MU[tid], num = 0.f;
      for (int i = 0; i < NI; ++i) { float d = V[(i * 16 + c) * 16 + hh] - m; num += R[i * 16 + c] * d * d; }
      SG[tid] = num / (rsum[c] + EPSF) + EPSF;
    }
    __syncthreads();
    if (tid < 16) {                          // a_out
      float cost = 0.f;
      for (int hh = 0; hh < 16; ++hh) cost += beta_u[tid] + 0.5f * __logf(SG[tid * 16 + hh]);
      cost *= rsum[tid];
      AO[tid] = 1.0f / (1.0f + __expf(-(LAMF * (beta_a[tid] - cost))));
    }
    __syncthreads();
    if (it < 2) {
      if (tid < NI) {                        // softmax(ln_ap) over c
        float lnap[16], mx = -1e30f;
        for (int c = 0; c < 16; ++c) {
          float s2 = 0.f;
          for (int hh = 0; hh < 16; ++hh) {
            float sg = SG[c * 16 + hh];
            float d = V[(tid * 16 + c) * 16 + hh] - MU[c * 16 + hh];
            s2 += -d * d / (2.0f * sg) - 0.5f * __logf(sg);
          }
          s2 += -8.0f * LN2PI + __logf(AO[c]);
          lnap[c] = s2; mx = fmaxf(mx, s2);
        }
        float se = 0.f;
        for (int c = 0; c < 16; ++c) { float e = __expf(lnap[c] - mx); lnap[c] = e; se += e; }
        float inv = 1.0f / se;
        for (int c = 0; c < 16; ++c) R[tid * 16 + c] = lnap[c] * inv;
      }
      __syncthreads();
    }
  }
  pose1[n * 256 + tid] = MU[tid];            // channel = c*16+h = tid
  if (tid < 16) a1[n * 16 + tid] = AO[tid];
}

// ---------------------------------------------------------------
// 5) class_caps: votes-on-the-fly + EM, one block per batch element.
// ---------------------------------------------------------------
__device__ __forceinline__ float cls_vote(const float* __restrict__ P,
                                          const float* __restrict__ W,
                                          int j, int c, int hh) {
  int i = j & 15, m = j >> 4;
  int p = hh >> 2, r = hh & 3;
  const float* ps = &P[j * 16 + p * 4];
  const float* ww = &W[(i * 10 + c) * 16 + r];
  float acc = ps[0] * ww[0] + ps[1] * ww[4] + ps[2] * ww[8] + ps[3] * ww[12];
  if (hh == 0)      acc += (float)(m / 6) * (1.0f / 6.0f);
  else if (hh == 1) acc += (float)(m % 6) * (1.0f / 6.0f);
  return acc;
}

__global__ void class_caps_em(const float* __restrict__ pose1, const float* __restrict__ a1,
                              const float* __restrict__ clsw,
                              const float* __restrict__ beta_u, const float* __restrict__ beta_a,
                              float* __restrict__ pose_c, float* __restrict__ a_c) {
  extern __shared__ float sm[];
  float* P    = sm;                  // 576*16
  float* W    = P + NCAP * 16;       // 2560
  float* AIn  = W + 2560;            // 576
  float* R    = AIn + NCAP;          // 576*10
  float* MU   = R + NCAP * 10;       // 160
  float* SG   = MU + 160;            // 160
  float* rsum = SG + 160;            // 10
  float* AO   = rsum + 10;           // 10

  int b = blockIdx.x, tid = threadIdx.x;
  for (int e = tid; e < NCAP * 16; e += 256) P[e] = pose1[b * NCAP * 16 + e];
  for (int e = tid; e < 2560; e += 256) W[e] = clsw[e];
  for (int e = tid; e < NCAP; e += 256) AIn[e] = a1[b * NCAP + e];
  for (int e = tid; e < NCAP * 10; e += 256) R[e] = 0.1f;
  __syncthreads();

  for (int it = 0; it < 3; ++it) {
    for (int j = tid; j < NCAP; j += 256) {
      float sum = 0.f;
      for (int c = 0; c < 10; ++c) { float t = R[j * 10 + c] * AIn[j]; R[j * 10 + c] = t; sum += t; }
      float inv = 1.0f / (sum + EPSF);
      for (int c = 0; c < 10; ++c) R[j * 10 + c] *= inv;
    }
    __syncthreads();
    if (tid < 10) rsum[tid] = 0.f;
    __syncthreads();
    {
      float part[10];
      for (int c = 0; c < 10; ++c) part[c] = 0.f;
      for (int j = tid; j < NCAP; j += 256)
        for (int c = 0; c < 10; ++c) part[c] += R[j * 10 + c];
      for (int c = 0; c < 10; ++c) atomicAdd(&rsum[c], part[c]);   // ds_add_f32
    }
    __syncthreads();
    if (tid < 160) {                          // mu
      int c = tid / 16, hh = tid & 15;
      float num = 0.f;
      for (int j = 0; j < NCAP; ++j) num += R[j * 10 + c] * cls_vote(P, W, j, c, hh);
      MU[tid] = num / (rsum[c] + EPSF);
    }
    __syncthreads();
    if (tid < 160) {                          // sigma_sq
      int c = tid / 16, hh = tid & 15;
      float m = MU[tid], num = 0.f;
      for (int j = 0; j < NCAP; ++j) { float d = cls_vote(P, W, j, c, hh) - m; num += R[j * 10 + c] * d * d; }
      SG[tid] = num / (rsum[c] + EPSF) + EPSF;
    }
    __syncthreads();
    if (tid < 10) {
      float cost = 0.f;
      for (int hh = 0; hh < 16; ++hh) cost += beta_u[tid] + 0.5f * __logf(SG[tid * 16 + hh]);
      cost *= rsum[tid];
      AO[tid] = 1.0f / (1.0f + __expf(-(LAMF * (beta_a[tid] - cost))));
    }
    __syncthreads();
    if (it < 2) {
      for (int j = tid; j < NCAP; j += 256) {
        float lnap[10], mx = -1e30f;
        for (int c = 0; c < 10; ++c) {
          float s2 = 0.f;
          for (int hh = 0; hh < 16; ++hh) {
            float sg = SG[c * 16 + hh];
            float d = cls_vote(P, W, j, c, hh) - MU[c * 16 + hh];
            s2 += -d * d / (2.0f * sg) - 0.5f * __logf(sg);
          }
          s2 += -8.0f * LN2PI + __logf(AO[c]);
          lnap[c] = s2; mx = fmaxf(mx, s2);
        }
        float se = 0.f;
        for (int c = 0; c < 10; ++c) { float e = __expf(lnap[c] - mx); lnap[c] = e; se += e; }
        float inv = 1.0f / se;
        for (int c = 0; c < 10; ++c) R[j * 10 + c] = lnap[c] * inv;
      }
      __syncthreads();
    }
  }
  if (tid < 160) pose_c[b * 160 + tid] = MU[tid];
  if (tid < 10)  a_c[b * 10 + tid] = AO[tid];
}

// ---------------------------------------------------------------
// 6) label-select pose -> z(64,160)
// ---------------------------------------------------------------
__global__ void select_z(const float* __restrict__ pose_c, const int* __restrict__ y,
                         float* __restrict__ z) {
  int t = blockIdx.x * blockDim.x + threadIdx.x;
  if (t >= BATCH * 160) return;
  int b = t / 160, k = t % 160, e = k / 16;
  z[t] = (y[b] == e) ? pose_c[t] : 0.0f;
}

// ---------------------------------------------------------------
// 7) decoder GEMMs: out[M,N] = act(A[M,K] @ W[N,K]^T + bias), WMMA f32
//    one wave per 16x16 output tile; ACT: 1=relu, 2=sigmoid
// ---------------------------------------------------------------
template <int ACT>
__global__ void gemm_wt_wmma(const float* __restrict__ A, const float* __restrict__ W,
                             const float* __restrict__ bias, float* __restrict__ out,
                             int N, int Kdim) {
  int nt = blockIdx.x, mt = blockIdx.y;
  int lane = threadIdx.x;
  int hf = lane >> 4, l = lane & 15;
  int mrow = mt * 16 + l;
  int ncol = nt * 16 + l;
  v8f acc = {};
  for (int k0 = 0; k0 < Kdim; k0 += 4) {
    int ka = k0 + hf * 2;
    v2f av, bv;
    av.x = A[mrow * Kdim + ka];     av.y = A[mrow * Kdim + ka + 1];
    bv.x = W[ncol * Kdim + ka];     bv.y = W[ncol * Kdim + ka + 1];
    acc = __builtin_amdgcn_wmma_f32_16x16x4_f32(false, av, false, bv,
                                                (short)0, acc, false, false);
  }
  int col = nt * 16 + l;
  float bb = bias[col];
  for (int v = 0; v < 8; ++v) {
    float val = acc[v] + bb;
    if (ACT == 1) val = fmaxf(val, 0.0f);
    if (ACT == 2) val = 1.0f / (1.0f + __expf(-val));
    out[(mt * 16 + v + hf * 8) * N + col] = val;
  }
}

// ---------------------------------------------------------------
extern "C" void kernel_launch(void* const* d_in, const int* in_sizes, int n_in,
                              void* d_out, int out_size, void* d_ws, size_t ws_size,
                              hipStream_t stream) {
  const float* x      = (const float*)d_in[0];
  const int*   y      = (const int*)  d_in[1];
  const float* c1w    = (const float*)d_in[2];
  const float* c1b    = (const float*)d_in[3];
  const float* ppw    = (const float*)d_in[4];
  const float* ppb    = (const float*)d_in[5];
  const float* paw    = (const float*)d_in[6];
  const float* pab    = (const float*)d_in[7];
  const float* ccw    = (const float*)d_in[8];
  const float* cc_bu  = (const float*)d_in[9];
  const float* cc_ba  = (const float*)d_in[10];
  const float* clsw   = (const float*)d_in[11];
  const float* cls_bu = (const float*)d_in[12];
  const float* cls_ba = (const float*)d_in[13];
  const float* w1     = (const float*)d_in[14];
  const float* b1     = (const float*)d_in[15];
  const float* w2     = (const float*)d_in[16];
  const float* b2     = (const float*)d_in[17];

  float* out = (float*)d_out;          // [a_c 64*10 | recon 64*784]
  float* ws  = (float*)d_ws;
  float* h     = ws;                   // 802816
  float* pose  = h     + 802816;       // 1605632 (NHWC 64,14,14,128)
  float* act   = pose  + 1605632;      // 100352  (NHWC 64,14,14,8)
  float* pose1 = act   + 100352;       // 589824  (64,6,6,256)
  float* a1    = pose1 + 589824;       // 36864   (64,6,6,16)
  float* poseC = a1    + 36864;        // 10240   (64,10,16)
  float* z     = poseC + 10240;        // 10240
  float* hdec  = z     + 10240;        // 32768

  conv1_relu<<<3136, 256, 0, stream>>>(x, c1w, c1b, h);
  pc_pose_wmma<<<dim3(8, 784), 32, 0, stream>>>(h, ppw, ppb, pose);
  pc_act<<<392, 256, 0, stream>>>(h, paw, pab, act);

  size_t sm_c = (size_t)(NI * 256 + NI * 16 + NI + 16 + 256 + 256 + 16) * sizeof(float);
  conv_caps_em<<<2304, 256, sm_c, stream>>>(pose, act, ccw, cc_bu, cc_ba, pose1, a1);

  size_t sm_d = (size_t)(NCAP * 16 + 2560 + NCAP + NCAP * 10 + 160 + 160 + 10 + 10) * sizeof(float);
  class_caps_em<<<64, 256, sm_d, stream>>>(pose1, a1, clsw, cls_bu, cls_ba, poseC, out);

  select_z<<<40, 256, 0, stream>>>(poseC, y, z);
  gemm_wt_wmma<1><<<dim3(32, 4), 32, 0, stream>>>(z, w1, b1, hdec, 512, 160);
  gemm_wt_wmma<2><<<dim3(49, 4), 32, 0, stream>>>(hdec, w2, b2, out + 640, 784, 512);
}